// StackedFlow_81913616269685
// MI455X (gfx1250) — compile-verified
//
#include <hip/hip_runtime.h>
#include <hip/hip_bf16.h>

// ---------------------------------------------------------------------------
// StackedFlow on MI455X (gfx1250), fp32 WMMA path.
//
// All dense math is routed through V_WMMA_F32_16X16X4_F32 (wave32, one wave
// per 16x16 C tile).  The log-space jacobian propagation is rewritten as
//   logsumexp(log(Wdiag)+g) = log( Wdiag @ exp(g - m) ) + m
// (valid because the diagonal blocks are clamped positive), turning the
// transcendental-heavy reduction into batched WMMA GEMMs.
// Epilogues use native gfx1250 transcendentals (v_tanh_f32 / v_exp_f32 /
// v_log_f32) instead of inlined libm ladders.
// ---------------------------------------------------------------------------

typedef float v2f __attribute__((ext_vector_type(2)));
typedef float v8f __attribute__((ext_vector_type(8)));

#define Bsz   256
#define Dd    16
#define Mm    128
#define HID   2048   // Mm * Dd

__device__ __forceinline__ v8f wmma_f32_k4(v2f a, v2f b, v8f c) {
  // D = A(16x4) * B(4x16) + C, fp32, wave32.  8-arg form:
  // (neg_a, A, neg_b, B, c_mod, C, reuse_a, reuse_b)
  return __builtin_amdgcn_wmma_f32_16x16x4_f32(
      false, a, false, b, (short)0, c, false, false);
}

// One wave computes a 16x16 tile of  C = A(MxK) @ W(NxK)^T   (row-major A, W).
// A-operand layout (ISA 7.12.2, 32-bit A 16x4): lane = M (mod 16), half-wave
// selects K pair.  B-operand mirrored (lane = N, half-wave selects K pair).
__device__ __forceinline__ void gemm_tile(const float* __restrict__ A, int lda,
                                          const float* __restrict__ W, int ldw,
                                          int m0, int n0, int K, v8f& acc) {
  const int lane = threadIdx.x & 31;
  const int r    = lane & 15;
  const int kh   = (lane >> 4) << 1;         // 0 or 2
  const float* arow = A + (size_t)(m0 + r) * lda;
  const float* wrow = W + (size_t)(n0 + r) * ldw;
  for (int k = 0; k < K; k += 4) {
    v2f a, b;
    a.x = arow[k + kh];  a.y = arow[k + kh + 1];
    b.x = wrow[k + kh];  b.y = wrow[k + kh + 1];
    acc = wmma_f32_k4(a, b, acc);
  }
}

// ---- native-transcendental epilogue helpers -------------------------------
__device__ __forceinline__ float fast_tanh(float x) {
#if defined(__has_builtin)
#if __has_builtin(__builtin_amdgcn_tanhf)
  return __builtin_amdgcn_tanhf(x);          // v_tanh_f32
#elif __has_builtin(__builtin_amdgcn_tanh_f32)
  return __builtin_amdgcn_tanh_f32(x);       // v_tanh_f32
#else
  return tanhf(x);
#endif
#else
  return tanhf(x);
#endif
}

__device__ __forceinline__ float softplus_f(float x) {   // log(1+e^x), stable
  // e = exp(-|x|) in (0,1]; when 1+e rounds to 1 the term is < 2^-24 and is
  // negligible against the max(x,0) part, so hardware v_log_f32 is safe.
  return fmaxf(x, 0.0f) + __logf(1.0f + __expf(-fabsf(x)));
}
__device__ __forceinline__ float log_abs_dtanh(float h) { // log|tanh'(h)|
  return 2.0f * (0.6931471805599453f - h - softplus_f(-2.0f * h));
}

// ---- layer 0: h = X @ W0^T + b0 ; act = tanh(h); grad = log(W0diag) + lga --
// X:(256x16)  W0:(2048x16)  -> act, grad : (256x2048)
__global__ void k_layer0(const float* __restrict__ X, const float* __restrict__ W0,
                         const float* __restrict__ b0,
                         float* __restrict__ act, float* __restrict__ grad) {
  const int wave = (blockIdx.x * blockDim.x + threadIdx.x) >> 5;
  const int mt = wave >> 7;          // 16 M-tiles
  const int nt = wave & 127;         // 128 N-tiles
  if (mt >= Bsz / 16) return;
  v8f acc = {};
  gemm_tile(X, Dd, W0, Dd, mt * 16, nt * 16, Dd, acc);

  const int lane = threadIdx.x & 31;
  const int n = nt * 16 + (lane & 15);
  const int rowoff = (lane >> 4) * 8;
  const float ldiag = __logf(W0[n * Dd + (n >> 7)]);   // log W0[n][n/128]
  const float bias = b0[n];
  for (int v = 0; v < 8; ++v) {
    const int m = mt * 16 + v + rowoff;
    const float h = acc[v] + bias;
    act[m * HID + n]  = fast_tanh(h);
    grad[m * HID + n] = ldiag + log_abs_dtanh(h);
  }
}

// ---- per-(b,k) row of 128: m = max(g), E = exp(g - m) ----------------------
// grad laid out so that (b,k) row is contiguous: index = (b*16+k)*128 + i
__global__ void k_grad_prep(const float* __restrict__ grad,
                            float* __restrict__ mrow, float* __restrict__ E) {
  const int bk = blockIdx.x;          // 0..4095
  const int i  = threadIdx.x;         // 0..127
  const int idx = bk * Mm + i;
  const float g = grad[idx];
  __shared__ float red[Mm];
  red[i] = g;
  __syncthreads();
  for (int s = 64; s > 0; s >>= 1) {
    if (i < s) red[i] = fmaxf(red[i], red[i + s]);
    __syncthreads();
  }
  const float mx = red[0];
  E[idx] = __expf(g - mx);
  if (i == 0) mrow[bk] = mx;
}

// ---- layer 1: h = act0 @ W1^T + b1 ; act1 = tanh(h); lga stored ------------
__global__ void k_layer1(const float* __restrict__ A, const float* __restrict__ W1,
                         const float* __restrict__ b1,
                         float* __restrict__ act, float* __restrict__ lga) {
  const int wave = (blockIdx.x * blockDim.x + threadIdx.x) >> 5;
  const int mt = wave >> 7;
  const int nt = wave & 127;
  if (mt >= Bsz / 16) return;
  v8f acc = {};
  gemm_tile(A, HID, W1, HID, mt * 16, nt * 16, HID, acc);

  const int lane = threadIdx.x & 31;
  const int n = nt * 16 + (lane & 15);
  const int rowoff = (lane >> 4) * 8;
  const float bias = b1[n];
  for (int v = 0; v < 8; ++v) {
    const int m = mt * 16 + v + rowoff;
    const float h = acc[v] + bias;
    act[m * HID + n] = fast_tanh(h);
    lga[m * HID + n] = log_abs_dtanh(h);
  }
}

// ---- grad update via batched GEMM on diagonal blocks of W1 -----------------
// For k-slice: S(256x128) = E_k(256x128) @ W1diag_k(128x128)^T
// grad_new[b][k*128+o] = log(S) + m[b,k] + lga[b][k*128+o]
__global__ void k_grad_gemm(const float* __restrict__ E, const float* __restrict__ W1,
                            const float* __restrict__ lga, const float* __restrict__ mrow,
                            float* __restrict__ grad) {
  const int wave = (blockIdx.x * blockDim.x + threadIdx.x) >> 5;
  const int ks  = wave >> 7;          // 16 k-slices
  const int rem = wave & 127;
  const int mt  = rem >> 3;           // 16 M-tiles (batch)
  const int nt  = rem & 7;            // 8 N-tiles (128 outputs)
  if (ks >= Dd) return;
  const float* Ak = E  + (size_t)ks * Mm;                         // lda = 2048
  const float* Wk = W1 + (size_t)ks * Mm * HID + (size_t)ks * Mm; // ldw = 2048
  v8f acc = {};
  gemm_tile(Ak, HID, Wk, HID, mt * 16, nt * 16, Mm, acc);

  const int lane = threadIdx.x & 31;
  const int o = nt * 16 + (lane & 15);
  const int rowoff = (lane >> 4) * 8;
  for (int v = 0; v < 8; ++v) {
    const int b = mt * 16 + v + rowoff;
    const float s = acc[v];
    grad[b * HID + ks * Mm + o] =
        __logf(s) + mrow[b * Dd + ks] + lga[b * HID + ks * Mm + o];
  }
}

// ---- final linear layer: out(256x16) = act1 @ W2(16x2048)^T ----------------
__global__ void k_out_gemm(const float* __restrict__ A, const float* __restrict__ W2,
                           float* __restrict__ out) {
  const int wave = threadIdx.x >> 5;     // 16 waves = 16 M-tiles, 1 N-tile
  if (wave >= Bsz / 16) return;
  v8f acc = {};
  gemm_tile(A, HID, W2, HID, wave * 16, 0, HID, acc);
  const int lane = threadIdx.x & 31;
  const int n = lane & 15;
  const int rowoff = (lane >> 4) * 8;
  for (int v = 0; v < 8; ++v) {
    const int m = wave * 16 + v + rowoff;
    out[m * Dd + n] = acc[v];
  }
}

// ---- log-det: per sample, 16 dots of length 128 against W2 diag blocks -----
__global__ void k_logdet(const float* __restrict__ E, const float* __restrict__ mrow,
                         const float* __restrict__ W2, float* __restrict__ ld,
                         int accumulate) {
  const int b = blockIdx.x * blockDim.x + threadIdx.x;
  if (b >= Bsz) return;
  float tot = 0.0f;
  for (int k = 0; k < Dd; ++k) {
    const float* w = W2 + k * HID + k * Mm;   // W2[k][k*128 + i]
    const float* e = E + b * HID + k * Mm;
    float s = 0.0f;
    for (int i = 0; i < Mm; ++i) s = fmaf(w[i], e[i], s);
    tot += __logf(s) + mrow[b * Dd + k];
  }
  if (accumulate) ld[b] += tot; else ld[b] = tot;
}

// ---------------------------------------------------------------------------
static void run_flow(const float* X, const float* W0, const float* W1,
                     const float* W2, const float* b0, const float* b1,
                     float* x_out, float* ld_out, int accumulate,
                     float* act0, float* act1, float* lga, float* grad,
                     float* E, float* mrow, hipStream_t stream) {
  // 2048 tiles -> 2048 waves -> 65536 threads
  k_layer0   <<<256, 256, 0, stream>>>(X, W0, b0, act0, grad);
  k_grad_prep<<<Bsz * Dd, Mm, 0, stream>>>(grad, mrow, E);
  k_layer1   <<<256, 256, 0, stream>>>(act0, W1, b1, act1, lga);
  k_grad_gemm<<<256, 256, 0, stream>>>(E, W1, lga, mrow, grad);
  k_grad_prep<<<Bsz * Dd, Mm, 0, stream>>>(grad, mrow, E);
  k_out_gemm <<<1, 512, 0, stream>>>(act1, W2, x_out);
  k_logdet   <<<1, 256, 0, stream>>>(E, mrow, W2, ld_out, accumulate);
}

extern "C" void kernel_launch(void* const* d_in, const int* in_sizes, int n_in,
                              void* d_out, int out_size, void* d_ws, size_t ws_size,
                              hipStream_t stream) {
  const float* inp   = (const float*)d_in[0];
  const float* f0_W0 = (const float*)d_in[1];
  const float* f0_W1 = (const float*)d_in[2];
  const float* f0_W2 = (const float*)d_in[3];
  const float* f0_b0 = (const float*)d_in[4];
  const float* f0_b1 = (const float*)d_in[5];
  const float* f1_W0 = (const float*)d_in[6];
  const float* f1_W1 = (const float*)d_in[7];
  const float* f1_W2 = (const float*)d_in[8];
  const float* f1_b0 = (const float*)d_in[9];
  const float* f1_b1 = (const float*)d_in[10];

  float* out = (float*)d_out;          // (256 x 16) flattened
  float* ld  = out + Bsz * Dd;         // (256,)

  // workspace layout (floats): ~10.5 MB total
  float* ws   = (float*)d_ws;
  float* act0 = ws;                    // 256*2048
  float* act1 = act0 + Bsz * HID;
  float* lga  = act1 + Bsz * HID;
  float* grad = lga  + Bsz * HID;
  float* E    = grad + Bsz * HID;
  float* mrow = E    + Bsz * HID;      // 256*16
  float* xmid = mrow + Bsz * Dd;       // 256*16

  run_flow(inp,  f0_W0, f0_W1, f0_W2, f0_b0, f0_b1, xmid, ld, /*accum=*/0,
           act0, act1, lga, grad, E, mrow, stream);
  run_flow(xmid, f1_W0, f1_W1, f1_W2, f1_b0, f1_b1, out,  ld, /*accum=*/1,
           act0, act1, lga, grad, E, mrow, stream);
}